// MeanSquaredError224HM_59287728554152
// MI455X (gfx1250) — compile-verified
//
#include <hip/hip_runtime.h>
#include <hip/hip_bf16.h>

typedef __attribute__((ext_vector_type(2))) float v2f;
typedef __attribute__((ext_vector_type(8))) float v8f;

#define COLN 224
#define RAD 4
#define TH_MAIN 256
#define CHUNKS 7

// ---------------------------------------------------------------------------
// Kernel 1: per-joint prep. Computes visibility mask, window origin (xs,ys),
// and the reflect-folded, peak-normalized 1-D Gaussian weights (9 each axis).
// tt(y,x) = wyn(y)*wxn(x) exactly reproduces blur(delta)/max after symmetric
// padding, because symmetric padding + separable kernel stays separable.
// ---------------------------------------------------------------------------
__global__ void prep_joints_kernel(const float* __restrict__ t,
                                   const int* __restrict__ v, int nJ,
                                   int* __restrict__ maskArr,
                                   int* __restrict__ xsArr,
                                   int* __restrict__ ysArr,
                                   float* __restrict__ wxn,
                                   float* __restrict__ wyn) {
  int j = blockIdx.x * blockDim.x + threadIdx.x;
  if (j >= nJ) return;

  float tx = t[2 * j + 0] * (float)COLN;
  float ty = t[2 * j + 1] * (float)COLN;
  int xi = (int)floorf(tx);
  int yi = (int)floorf(ty);
  bool visible = (v[2 * j + 0] == 1);
  bool inb = (xi >= 0) && (xi <= COLN - 1) && (yi >= 0) && (yi <= COLN - 1);
  int mask = (visible && inb) ? 1 : 0;
  maskArr[j] = mask;

  // normalized truncated 1-D Gaussian, sigma=1, radius 4
  float g[2 * RAD + 1];
  float gs = 0.f;
  for (int k = 0; k <= 2 * RAD; k++) {
    float d = (float)(k - RAD);
    g[k] = expf(-0.5f * d * d);
    gs += g[k];
  }
  for (int k = 0; k <= 2 * RAD; k++) g[k] /= gs;

  int xs = xi - RAD; if (xs < 0) xs = 0;
  int ys = yi - RAD; if (ys < 0) ys = 0;
  xsArr[j] = xs;
  ysArr[j] = ys;

  float wxv[9], wyv[9], mxx = 0.f, mxy = 0.f;
  for (int i = 0; i < 9; i++) {
    // x axis
    int x = xs + i;
    float w = 0.f;
    if (x <= COLN - 1) {
      for (int k = 0; k <= 2 * RAD; k++) {
        int m = x + k - RAD;
        int mr = (m < 0) ? (-1 - m) : ((m > COLN - 1) ? (2 * COLN - 1 - m) : m);
        if (mr == xi) w += g[k];
      }
    }
    wxv[i] = w; mxx = fmaxf(mxx, w);
    // y axis
    int y = ys + i;
    float wy = 0.f;
    if (y <= COLN - 1) {
      for (int k = 0; k <= 2 * RAD; k++) {
        int m = y + k - RAD;
        int mr = (m < 0) ? (-1 - m) : ((m > COLN - 1) ? (2 * COLN - 1 - m) : m);
        if (mr == yi) wy += g[k];
      }
    }
    wyv[i] = wy; mxy = fmaxf(mxy, wy);
  }
  for (int i = 0; i < 9; i++) {
    wxn[j * 9 + i] = (mxx > 0.f) ? (wxv[i] / mxx) : 0.f;
    wyn[j * 9 + i] = (mxy > 0.f) ? (wyv[i] / mxy) : 0.f;
  }
}

// ---------------------------------------------------------------------------
// Kernel 2: streaming sum of h^2 over masked maps (the bandwidth-bound part).
// grid = (nMaps, CHUNKS), block = 256, 7 float4 per thread. Wave-level
// reduction done with V_WMMA_F32_16X16X4_F32 against an all-ones B matrix:
// D[m][n] = s_m + s_{16+m}, so sum(d[0..7]) + shfl_xor(16) = wave total.
// ---------------------------------------------------------------------------
__global__ void sumsq_kernel(const float* __restrict__ h,
                             const int* __restrict__ maskArr,
                             float* __restrict__ partial,
                             int f4PerMap, int iters) {
  const int m = blockIdx.x;
  const int chunk = blockIdx.y;
  const int tid = threadIdx.x;
  const float4* __restrict__ hv =
      (const float4*)h + (size_t)m * (size_t)f4PerMap;

  float acc = 0.f;
  if (maskArr[m]) {
    int base = chunk * iters * TH_MAIN + tid;
    #pragma unroll
    for (int i = 0; i < CHUNKS; i++) {
      if (i >= iters) break;
      int idx = base + i * TH_MAIN;
      if (idx < f4PerMap) {
        if (idx + TH_MAIN < f4PerMap)
          __builtin_prefetch(&hv[idx + TH_MAIN], 0, 0);  // global_prefetch_b8
        float4 x = hv[idx];
        acc = fmaf(x.x, x.x, acc);
        acc = fmaf(x.y, x.y, acc);
        acc = fmaf(x.z, x.z, acc);
        acc = fmaf(x.w, x.w, acc);
      }
    }
  }

  __shared__ float sdata[TH_MAIN];
  sdata[tid] = acc;
  __syncthreads();

  if (tid < 32) {  // whole wave 0 -> EXEC all-ones inside this branch
    float s = 0.f;
    #pragma unroll
    for (int w = 0; w < TH_MAIN / 32; w++) s += sdata[tid + 32 * w];

    v2f a; a.x = s;   a.y = 0.f;   // A: 16x4, per-lane partial in K-slot 0/2
    v2f b; b.x = 1.f; b.y = 1.f;   // B: 4x16 all ones
    v8f c = {};
    v8f d = __builtin_amdgcn_wmma_f32_16x16x4_f32(
        false, a, false, b, (short)0, c, false, false);
    float tsum = d[0] + d[1] + d[2] + d[3] + d[4] + d[5] + d[6] + d[7];
    tsum += __shfl_xor(tsum, 16, 32);
    if (tid == 0) partial[(size_t)m * gridDim.y + chunk] = tsum;
  }
}

// ---------------------------------------------------------------------------
// Kernel 3: per-joint 9x9 window correction: (h-tt)^2 - h^2 = tt^2 - 2*h*tt.
// One block of 128 threads per joint; thread i<81 covers one window element.
// Zero weights (outside folded support / outside image) contribute nothing
// and guard the h read against out-of-map access.
// ---------------------------------------------------------------------------
__global__ void corr_kernel(const float* __restrict__ h,
                            const int* __restrict__ maskArr,
                            const int* __restrict__ xsArr,
                            const int* __restrict__ ysArr,
                            const float* __restrict__ wxn,
                            const float* __restrict__ wyn,
                            float* __restrict__ corr, int mapElems) {
  const int j = blockIdx.x;
  const int tid = threadIdx.x;
  float local = 0.f;
  if (maskArr[j] && tid < 81) {
    int iy = tid / 9, ix = tid % 9;
    float wy = wyn[j * 9 + iy];
    float wx = wxn[j * 9 + ix];
    float w = wy * wx;
    if (w > 0.f) {
      int y = ysArr[j] + iy;
      int x = xsArr[j] + ix;
      float hv = h[(size_t)j * (size_t)mapElems + (size_t)(y * COLN + x)];
      local = w * w - 2.0f * hv * w;
    }
  }
  __shared__ float sd[128];
  sd[tid] = local;
  __syncthreads();
  for (int s = 64; s > 0; s >>= 1) {
    if (tid < s) sd[tid] += sd[tid + s];
    __syncthreads();
  }
  if (tid == 0) corr[j] = sd[0];
}

// ---------------------------------------------------------------------------
// Kernel 4: deterministic final reduction (fixed-order strided sums + tree).
// ---------------------------------------------------------------------------
__global__ void finalize_kernel(const float* __restrict__ partial, int nPart,
                                const float* __restrict__ corr,
                                const int* __restrict__ maskArr, int nJ,
                                float* __restrict__ out) {
  __shared__ float sf[256];
  __shared__ int si[256];
  const int tid = threadIdx.x;
  float fa = 0.f;
  int ia = 0;
  for (int i = tid; i < nPart; i += 256) fa += partial[i];
  for (int i = tid; i < nJ; i += 256) {
    fa += corr[i];
    ia += maskArr[i];
  }
  sf[tid] = fa;
  si[tid] = ia;
  __syncthreads();
  for (int s = 128; s > 0; s >>= 1) {
    if (tid < s) { sf[tid] += sf[tid + s]; si[tid] += si[tid + s]; }
    __syncthreads();
  }
  if (tid == 0) {
    int c = si[0] < 1 ? 1 : si[0];
    out[0] = sf[0] / (float)c;
  }
}

extern "C" void kernel_launch(void* const* d_in, const int* in_sizes, int n_in,
                              void* d_out, int out_size, void* d_ws, size_t ws_size,
                              hipStream_t stream) {
  const float* h = (const float*)d_in[0];
  const float* t = (const float*)d_in[1];
  const int* v = (const int*)d_in[2];

  const int nJ = in_sizes[1] / 2;            // B*Nj = 1792
  const int mapElems = in_sizes[0] / nJ;     // 224*224 = 50176
  const int f4PerMap = mapElems / 4;         // 12544
  const int iters = (f4PerMap + CHUNKS * TH_MAIN - 1) / (CHUNKS * TH_MAIN);  // 7

  // Workspace layout (floats/ints), total ~208 KB
  float* ws = (float*)d_ws;
  float* partial = ws;                        // nJ * CHUNKS
  float* corr = partial + (size_t)nJ * CHUNKS;   // nJ
  int* maskArr = (int*)(corr + nJ);           // nJ
  int* xsArr = maskArr + nJ;                  // nJ
  int* ysArr = xsArr + nJ;                    // nJ
  float* wxn = (float*)(ysArr + nJ);          // nJ * 9
  float* wyn = wxn + (size_t)nJ * 9;          // nJ * 9

  prep_joints_kernel<<<(nJ + 255) / 256, 256, 0, stream>>>(
      t, v, nJ, maskArr, xsArr, ysArr, wxn, wyn);

  dim3 grid(nJ, CHUNKS);
  sumsq_kernel<<<grid, TH_MAIN, 0, stream>>>(h, maskArr, partial, f4PerMap, iters);

  corr_kernel<<<nJ, 128, 0, stream>>>(h, maskArr, xsArr, ysArr, wxn, wyn, corr,
                                      mapElems);

  finalize_kernel<<<1, 256, 0, stream>>>(partial, nJ * CHUNKS, corr, maskArr,
                                         nJ, (float*)d_out);
}